// HyperbolicVisitEncoder_23837068493265
// MI455X (gfx1250) — compile-verified
//
#include <hip/hip_runtime.h>
#include <hip/hip_bf16.h>
#include <math.h>
#include <stdint.h>

// ---- problem constants (from reference) ----
#define VOCAB      50000
#define DIM        128
#define NUM_VISITS 131072
#define MAX_CODES  16
#define EPS        1e-7f

#define WAVES_PER_BLOCK 4
#define BLOCK_THREADS   (WAVES_PER_BLOCK * 32)

typedef __attribute__((ext_vector_type(2))) float v2f;
typedef __attribute__((ext_vector_type(8))) float v8f;

// One wave (32 lanes, wave32) handles one visit.
// Phase 1: async-gather 16 embedding rows straight into LDS
//          (global_load_async_to_lds_b128, ASYNCcnt) -- no VGPR transit.
// Phase 2: masked sum via v_wmma_f32_16x16x4_f32 with the PAD mask carried
//          in the B matrix: C[d,n] = sum_k z_k[d] * m_k.
// Phase 3: hyperbolic log-map epilogue + coalesced float4 store.
__launch_bounds__(BLOCK_THREADS, 2)
__global__ void hve_kernel(const float* __restrict__ emb,
                           const int* __restrict__ codes,
                           float* __restrict__ out) {
  __shared__ float stage_all[WAVES_PER_BLOCK * MAX_CODES * DIM]; // 32 KB
  __shared__ float svec_all [WAVES_PER_BLOCK * DIM];             // 2 KB

  const int lane = threadIdx.x & 31;
  const int wave = threadIdx.x >> 5;
  const int v    = blockIdx.x * WAVES_PER_BLOCK + wave;

  float* stage = stage_all + wave * (MAX_CODES * DIM);
  float* svec  = svec_all  + wave * DIM;

  // ---- load this visit's 16 codes (lanes 0..15) ----
  int mycode = 0;
  if (lane < MAX_CODES) {
    mycode = codes[(size_t)v * MAX_CODES + lane];
  }

  // ---- async gather: 16 rows x 512B, one b128 per lane per row, direct
  //      memory->LDS (no VGPR round trip); mask handled later in B. ----
  float bm[MAX_CODES];
#pragma unroll
  for (int j = 0; j < MAX_CODES; ++j) {
    const int cj = __shfl(mycode, j);
    bm[j] = (cj != 0) ? 1.0f : 0.0f;
    const uint64_t gaddr =
        (uint64_t)(uintptr_t)(emb + (size_t)cj * DIM + lane * 4);
    // low 32 bits of a generic LDS pointer == LDS byte address
    const uint32_t laddr =
        (uint32_t)(uintptr_t)(stage + j * DIM + lane * 4);
    asm volatile("global_load_async_to_lds_b128 %0, %1, off"
                 :: "v"(laddr), "v"(gaddr)
                 : "memory");
  }

  const int h  = lane >> 4;   // lane half: selects K pair in A/B layouts
  const int al = lane & 15;

  // B matrix per K-step t (loop-invariant over chunks):
  //   lanes 0-15 : v0 = m[4t+0], v1 = m[4t+1]
  //   lanes 16-31: v0 = m[4t+2], v1 = m[4t+3]
  v2f bmat[4];
#pragma unroll
  for (int t = 0; t < 4; ++t) {
    bmat[t].x = h ? bm[4 * t + 2] : bm[4 * t + 0];
    bmat[t].y = h ? bm[4 * t + 3] : bm[4 * t + 1];
  }

  // wait for this wave's async gather to land in LDS
  asm volatile("s_wait_asynccnt 0x0" ::: "memory");

#if __has_builtin(__builtin_amdgcn_wmma_f32_16x16x4_f32)
  // A 16x4 f32 layout: lanes 0-15 -> (M=lane, K=0/1 in v0/v1),
  //                    lanes 16-31 -> (M=lane-16, K=2/3 in v0/v1).
  v8f acc[8];
#pragma unroll
  for (int c = 0; c < 8; ++c) {
    v8f a8 = {};
#pragma unroll
    for (int t = 0; t < 4; ++t) {
      const int r0 = 4 * t + 2 * h;    // rows 4t,4t+1 (h=0) / 4t+2,4t+3 (h=1)
      v2f a;
      a.x = stage[(r0    ) * DIM + 16 * c + al];
      a.y = stage[(r0 + 1) * DIM + 16 * c + al];
      a8 = __builtin_amdgcn_wmma_f32_16x16x4_f32(
          /*neg_a=*/false, a, /*neg_b=*/false, bmat[t],
          /*c_mod=*/(short)0, a8, /*reuse_a=*/false, /*reuse_b=*/false);
    }
    acc[c] = a8;
  }
  // ---- dump s[128] to LDS: reg r of lane-half h holds s[16c + 8h + r] ----
#pragma unroll
  for (int c = 0; c < 8; ++c) {
    if (al == 0) {
      *(float4*)(svec + 16 * c + 8 * h)     =
          make_float4(acc[c][0], acc[c][1], acc[c][2], acc[c][3]);
      *(float4*)(svec + 16 * c + 8 * h + 4) =
          make_float4(acc[c][4], acc[c][5], acc[c][6], acc[c][7]);
    }
  }
#else
  // Fallback (no WMMA): each lane sums its 4 dims across the 16 staged rows,
  // applying the PAD mask explicitly.
  {
    float4 s4 = make_float4(0.f, 0.f, 0.f, 0.f);
#pragma unroll
    for (int j = 0; j < MAX_CODES; ++j) {
      float4 z = *(const float4*)(stage + j * DIM + lane * 4);
      s4.x += bm[j] * z.x; s4.y += bm[j] * z.y;
      s4.z += bm[j] * z.z; s4.w += bm[j] * z.w;
    }
    *(float4*)(svec + 4 * lane) = s4;
  }
#endif
  __syncthreads();

  // ---- epilogue: s -> hyperbolic tangent-space vector ----
  float4 s4 = *(const float4*)(svec + 4 * lane);   // dims 4*lane..4*lane+3
  float  s0 = svec[0];

  float partial = s4.x * s4.x + s4.y * s4.y + s4.z * s4.z + s4.w * s4.w;
  if (lane == 0) partial -= s4.x * s4.x;           // exclude d=0
#pragma unroll
  for (int off = 16; off >= 1; off >>= 1)
    partial += __shfl_xor(partial, off);
  const float q = partial;                         // sum_{d>=1} s_d^2

  const float negLor = s0 * s0 - q;                // -lor
  const float denom  = sqrtf(fmaxf(negLor, EPS));  // sqrt(max(-lor, eps))
  const float x0     = fmaxf(s0 / denom, 1.0f + EPS);
  const float spn    = fmaxf(sqrtf(q) / denom, EPS);
  float coef = acoshf(x0) / (spn * denom);         // out_d = coef * s_d (d>=1)

  const int nonempty = __any(mycode != 0);         // lanes >=16 carry code 0
  if (!nonempty) coef = 0.0f;

  float4 o;
  o.x = coef * s4.x; o.y = coef * s4.y; o.z = coef * s4.z; o.w = coef * s4.w;
  if (lane == 0) o.x = 0.0f;                       // tangent[0] = 0

  *(float4*)(out + (size_t)v * DIM + 4 * lane) = o;
}

extern "C" void kernel_launch(void* const* d_in, const int* in_sizes, int n_in,
                              void* d_out, int out_size, void* d_ws, size_t ws_size,
                              hipStream_t stream) {
  (void)in_sizes; (void)n_in; (void)out_size; (void)d_ws; (void)ws_size;
  const float* emb   = (const float*)d_in[0];
  const int*   codes = (const int*)d_in[1];
  float*       out   = (float*)d_out;

  dim3 grid(NUM_VISITS / WAVES_PER_BLOCK);   // 32768 blocks, 1 visit per wave
  dim3 block(BLOCK_THREADS);                 // 128 threads = 4 wave32s
  hipLaunchKernelGGL(hve_kernel, grid, block, 0, stream, emb, codes, out);
}